// SSM_30786325578358
// MI455X (gfx1250) — compile-verified
//
#include <hip/hip_runtime.h>

// ---------------- problem constants ----------------
#define HID    64
#define TLEN   32768
#define NBATCH 256
#define CHUNK  64
#define NCHUNK (TLEN / CHUNK)     // 512
#define BT     16                 // batch columns per block
#define NBLK   (NBATCH / BT)      // 16 blocks

#define P1 65                     // LDS row pad (setup kernel)
#define P2 68                     // LDS row pad (main kernel, conflict-free wmma frag loads)

typedef float v2f __attribute__((ext_vector_type(2)));
typedef float v8f __attribute__((ext_vector_type(8)));
typedef int   v4i __attribute__((ext_vector_type(4)));

#if __has_builtin(__builtin_amdgcn_global_load_async_to_lds_b128) && \
    __has_builtin(__builtin_amdgcn_s_wait_asynccnt)
#define USE_ASYNC_LDS 1
typedef __attribute__((address_space(1))) v4i g_v4i;   // global int4
typedef __attribute__((address_space(3))) v4i l_v4i;   // LDS int4
#else
#define USE_ASYNC_LDS 0
#endif

// ws layout (floats): [0]=Ktri 64x64, [4096]=KB 64x64, [8192]=CA1 64x64, [12288]=AL 64x64
#define WS_KTRI 0
#define WS_KB   4096
#define WS_CA1  8192
#define WS_AL   12288

// ============================================================================
// Kernel 1: discretize + build chunk descriptors (tiny, one block, 64 threads)
// ============================================================================
__global__ __launch_bounds__(64) void ssm_setup(const float* __restrict__ A,
                                                const float* __restrict__ B,
                                                const float* __restrict__ C,
                                                float* __restrict__ ws) {
    // 3 reusable 64x65 f32 buffers (~50 KB) + small vectors
    __shared__ float s0[64 * P1];   // M  -> A_bar (persists)
    __shared__ float s1[64 * P1];   // I  -> P (squaring)
    __shared__ float s2[64 * P1];   // G  -> T (squaring temp)
    __shared__ float sBb[64], sCr[64], sKv[64], sW[64];

    const int r = threadIdx.x;      // this thread owns row r

    // M = I - 0.5A (s0), Inv = I (s1), G = I + 0.5A (s2)
    for (int j = 0; j < HID; ++j) {
        float a  = A[r * HID + j];
        float id = (r == j) ? 1.0f : 0.0f;
        s0[r * P1 + j] = id - 0.5f * a;
        s2[r * P1 + j] = id + 0.5f * a;
        s1[r * P1 + j] = id;
    }
    __syncthreads();

    // Gauss-Jordan: s1 <- (I - 0.5A)^-1   (diag-dominant, no pivoting)
    for (int k = 0; k < HID; ++k) {
        if (r == k) {
            float p = 1.0f / s0[k * P1 + k];
            for (int j = 0; j < HID; ++j) { s0[k * P1 + j] *= p; s1[k * P1 + j] *= p; }
        }
        __syncthreads();
        if (r != k) {
            float f = s0[r * P1 + k];
            for (int j = 0; j < HID; ++j) {
                s0[r * P1 + j] -= f * s0[k * P1 + j];
                s1[r * P1 + j] -= f * s1[k * P1 + j];
            }
        }
        __syncthreads();
    }

    // A_bar = Inv @ (I + 0.5A)  -> s0 ; B_bar = Inv @ (step*B), step = 1.0
    {
        float bb = 0.0f;
        for (int k = 0; k < HID; ++k) bb += s1[r * P1 + k] * B[k];
        float row[HID];
        for (int j = 0; j < HID; ++j) {
            float acc = 0.0f;
            for (int k = 0; k < HID; ++k) acc += s1[r * P1 + k] * s2[k * P1 + j];
            row[j] = acc;
        }
        __syncthreads();
        for (int j = 0; j < HID; ++j) s0[r * P1 + j] = row[j];  // A_bar
        sBb[r] = bb;
        sCr[r] = C[r];
    }
    __syncthreads();

    // Iterate c-row: kloc[t] = C A^t B ; CA1[t] = C A^{t+1}
    for (int t = 0; t < CHUNK; ++t) {
        float acc = 0.0f;
        for (int k = 0; k < HID; ++k) acc += sCr[k] * s0[k * P1 + r];
        if (r == 0) {
            float s = 0.0f;
            for (int k = 0; k < HID; ++k) s += sCr[k] * sBb[k];
            sKv[t] = s;
        }
        __syncthreads();
        sCr[r] = acc;
        ws[WS_CA1 + t * HID + r] = acc;
        __syncthreads();
    }

    // Iterate w: KB[:, 63-t] = A^t B
    sW[r] = sBb[r];
    __syncthreads();
    for (int t = 0; t < CHUNK; ++t) {
        ws[WS_KB + r * HID + (CHUNK - 1 - t)] = sW[r];
        float acc = 0.0f;
        for (int k = 0; k < HID; ++k) acc += s0[r * P1 + k] * sW[k];
        __syncthreads();
        sW[r] = acc;
        __syncthreads();
    }

    // AL = A_bar^64 by 6 squarings (s1 = P, s2 = T)
    for (int j = 0; j < HID; ++j) s1[r * P1 + j] = s0[r * P1 + j];
    __syncthreads();
    for (int s = 0; s < 6; ++s) {
        for (int j = 0; j < HID; ++j) {
            float acc = 0.0f;
            for (int k = 0; k < HID; ++k) acc += s1[r * P1 + k] * s1[k * P1 + j];
            s2[r * P1 + j] = acc;
        }
        __syncthreads();
        for (int j = 0; j < HID; ++j) s1[r * P1 + j] = s2[r * P1 + j];
        __syncthreads();
    }
    for (int j = 0; j < HID; ++j) ws[WS_AL + r * HID + j] = s1[r * P1 + j];

    // Ktri[t][i] = (i<=t) ? kloc[t-i] : 0   (thread r = row t)
    __syncthreads();
    for (int i = 0; i < CHUNK; ++i)
        ws[WS_KTRI + r * HID + i] = (i <= r) ? sKv[r - i] : 0.0f;
}

// ============================================================================
// 16x16 f32 WMMA tile: acc += W[m0:m0+16, 0:64] @ Xt[:, 0:64]^T
//   W  : LDS, row-major [64][P2]
//   Xt : LDS, [16][P2] holding operand transposed: Xt[n][k] = X[k][n]
// A frag (16x4 f32): lanes 0-15 rows M, VGPR0=K0,VGPR1=K1; lanes 16-31 K2,K3.
// B frag (4x16 f32): VGPR0 rows K0/K2 (lane halves), VGPR1 rows K1/K3.
// ============================================================================
__device__ __forceinline__ v8f gemm64_tile(const float* __restrict__ W,
                                           const float* __restrict__ Xt,
                                           int m0, v8f acc) {
    const int lane = threadIdx.x & 31;
    const int half = lane >> 4;       // 0/1 -> K pair offset
    const int r    = lane & 15;       // A row within tile / B column
    const float* wrow = W  + (m0 + r) * P2 + 2 * half;
    const float* xrow = Xt + r * P2 + 2 * half;
#pragma unroll
    for (int kk = 0; kk < HID; kk += 4) {
        v2f a = *(const v2f*)(wrow + kk);
        v2f b = *(const v2f*)(xrow + kk);
        acc = __builtin_amdgcn_wmma_f32_16x16x4_f32(false, a, false, b,
                                                    (short)0, acc, false, false);
    }
    return acc;
}

// ============================================================================
// Kernel 2: fused chunked scan. 16 blocks x 128 threads (4 waves).
// Block = 16 batch columns; wave = one 16-row M tile of the 64-row outputs.
// Per chunk:  Y = Ktri@U + CA1@X ; X' = KB@U + AL@X
// U is double-buffered; chunk c+1 is fetched with GLOBAL_LOAD_ASYNC_TO_LDS
// (ASYNCcnt) while chunk c is consumed by the WMMA chain.
// ============================================================================
__global__ __launch_bounds__(128) void ssm_main(const float* __restrict__ u,
                                                const float* __restrict__ ws,
                                                float* __restrict__ y) {
    __shared__ float sK [HID * P2];
    __shared__ float sKB[HID * P2];
    __shared__ float sCA[HID * P2];
    __shared__ float sAL[HID * P2];
    __shared__ float sUd[2][BT * P2];   // U transposed, double-buffered: sUd[i][b][t]
    __shared__ float sX [BT * P2];      // X transposed: sX[b][h]

    const int tid  = threadIdx.x;
    const int wave = tid >> 5;
    const int lane = tid & 31;
    const int half = lane >> 4;
    const int r    = lane & 15;
    const int m0   = wave * 16;
    const int b0   = blockIdx.x * BT;

    // stage the four 64x64 descriptor matrices with pad-P2 rows
    for (int i = tid; i < HID * HID; i += 128) {
        int rr = i >> 6, cc = i & 63;
        sK [rr * P2 + cc] = ws[WS_KTRI + i];
        sKB[rr * P2 + cc] = ws[WS_KB   + i];
        sCA[rr * P2 + cc] = ws[WS_CA1  + i];
        sAL[rr * P2 + cc] = ws[WS_AL   + i];
    }
    for (int i = tid; i < BT * P2; i += 128) sX[i] = 0.0f;  // x_{-1} = 0
    __syncthreads();

    const int lb = tid >> 3;            // 0..15 : batch column this thread stages
    const int lt = (tid & 7) * 8;       // 0..56 : 8 contiguous times (two 16B blocks)
    const float* usrc = u + (size_t)(b0 + lb) * TLEN + lt;

#if USE_ASYNC_LDS
    // per-thread async issue: two B128 ops move this thread's 8 floats
    auto issue_async = [&](int c) {
        const float* src = usrc + c * CHUNK;
        float* dst = &sUd[c & 1][lb * P2 + lt];
        g_v4i* gs = (g_v4i*)src;
        l_v4i* ls = (l_v4i*)dst;
        __builtin_amdgcn_global_load_async_to_lds_b128(gs, ls, /*offset=*/0,  /*cpol=*/0);
        __builtin_amdgcn_global_load_async_to_lds_b128(gs, ls, /*offset=*/16, /*cpol=*/0);
    };
    issue_async(0);                      // prologue: chunk 0 in flight
#endif

    for (int c = 0; c < NCHUNK; ++c) {
        const int cur = c & 1;

#if USE_ASYNC_LDS
        if (c + 1 < NCHUNK) {
            issue_async(c + 1);                       // overlap next chunk
            __builtin_amdgcn_s_wait_asynccnt(2);      // in-order: chunk c landed
        } else {
            __builtin_amdgcn_s_wait_asynccnt(0);
        }
#else
        {   // synchronous fallback staging
            const float* src = usrc + c * CHUNK;
            float4 lo = *(const float4*)(src);
            float4 hi = *(const float4*)(src + 4);
            float* dst = &sUd[cur][lb * P2 + lt];
            *(float4*)(dst)     = lo;
            *(float4*)(dst + 4) = hi;
            if (c + 1 < NCHUNK)
                __builtin_prefetch(src + CHUNK, 0, 3);   // -> global_prefetch_b8
        }
#endif
        __syncthreads();   // U[c] visible block-wide; X' of c-1 visible

        // ---- four accumulated WMMA GEMMs ----
        const float* sUc = sUd[cur];
        v8f yacc = {};
        v8f xacc = {};
        yacc = gemm64_tile(sK,  sUc, m0, yacc);   // intra-chunk triangular conv
        yacc = gemm64_tile(sCA, sX,  m0, yacc);   // state -> output broadcast
        xacc = gemm64_tile(sKB, sUc, m0, xacc);   // input -> next state
        xacc = gemm64_tile(sAL, sX,  m0, xacc);   // state propagation
        __syncthreads();   // all reads of sX / sU done

        // ---- commit next state (transposed layout) ----
#pragma unroll
        for (int v = 0; v < 8; ++v)
            sX[r * P2 + m0 + 8 * half + v] = xacc[v];

        // ---- store y: lane r = batch col, rows are 8 consecutive times ----
        float* ydst = y + (size_t)(b0 + r) * TLEN + c * CHUNK + m0 + 8 * half;
        float4 y0 = {yacc[0], yacc[1], yacc[2], yacc[3]};
        float4 y1 = {yacc[4], yacc[5], yacc[6], yacc[7]};
        *(float4*)(ydst)     = y0;
        *(float4*)(ydst + 4) = y1;
    }
}

// ============================================================================
extern "C" void kernel_launch(void* const* d_in, const int* in_sizes, int n_in,
                              void* d_out, int out_size, void* d_ws, size_t ws_size,
                              hipStream_t stream) {
    const float* u = (const float*)d_in[0];   // (256,1,32768)
    const float* A = (const float*)d_in[1];   // (64,64)
    const float* B = (const float*)d_in[2];   // (64,1)
    const float* C = (const float*)d_in[3];   // (1,64)
    float* y  = (float*)d_out;                // (256,1,32768)
    float* ws = (float*)d_ws;                 // needs 16384 floats (64 KB)

    ssm_setup<<<1, 64, 0, stream>>>(A, B, C, ws);
    ssm_main<<<NBLK, 128, 0, stream>>>(u, ws, y);
}